// hd_classifier_73581379715550
// MI455X (gfx1250) — compile-verified
//
#include <hip/hip_runtime.h>

#define NCLS 100
#define DDIM 10000
#define NROW 16384
#define CTILES 7            // ceil(100/16) class tiles of 16
#define CH 64               // columns per update workgroup
#define UPD_THREADS 256
#define ROWGROUPS (UPD_THREADS / CH)   // 4 row-groups sharing one LDS tile

typedef __attribute__((ext_vector_type(2))) float v2f;
typedef __attribute__((ext_vector_type(8))) float v8f;

// ---------------------------------------------------------------------------
// Kernel 1: scores = encoded @ class_hvs.T using exact-fp32 WMMA 16x16x4.
// One wave -> 16 rows x 112 cols (7 tiles). 8 waves/block -> 128 rows/block.
// A fragment loaded once per K-step and reused across all 7 B tiles.
// ---------------------------------------------------------------------------
__global__ __launch_bounds__(256)
void gemm_scores_kernel(const float* __restrict__ enc,
                        const float* __restrict__ hvs,
                        float* __restrict__ scores)
{
    const int wave = threadIdx.x >> 5;
    const int lane = threadIdx.x & 31;
    const int half = lane >> 4;      // 0: K=0,1   1: K=2,3  (ISA 16x4 f32 layout)
    const int l16  = lane & 15;
    const int rowBase = (blockIdx.x * 8 + wave) * 16;

    const float* aptr = enc + (size_t)(rowBase + l16) * DDIM + 2 * half;

    const float* bptr[CTILES];
#pragma unroll
    for (int t = 0; t < CTILES; ++t) {
        int c = t * 16 + l16;
        if (c >= NCLS) c = NCLS - 1;     // clamp: in-bounds garbage, never stored
        bptr[t] = hvs + (size_t)c * DDIM + 2 * half;
    }

    v8f acc[CTILES] = {};

#pragma unroll 16
    for (int k0 = 0; k0 < DDIM; k0 += 4) {
        if ((k0 & 63) == 0)
            __builtin_prefetch(aptr + k0 + 512, 0, 1);   // global_prefetch_b8
        v2f a = *(const v2f*)(aptr + k0);
#pragma unroll
        for (int t = 0; t < CTILES; ++t) {
            v2f b = *(const v2f*)(bptr[t] + k0);
            acc[t] = __builtin_amdgcn_wmma_f32_16x16x4_f32(
                false, a, false, b, (short)0, acc[t], false, false);
        }
    }

    // C/D layout: VGPR j -> row (j + 8*half), lanes 0-15 carry N=l16
#pragma unroll
    for (int t = 0; t < CTILES; ++t) {
        int col = t * 16 + l16;
        if (col < NCLS) {
#pragma unroll
            for (int j = 0; j < 8; ++j) {
                int row = rowBase + j + 8 * half;
                scores[(size_t)row * NCLS + col] = acc[t][j];
            }
        }
    }
}

// ---------------------------------------------------------------------------
// Kernel 2: per-row argmax (first-max tie-break, like jnp.argmax) + mis mask.
// meta[2i] = target if misclassified else -1 ; meta[2i+1] = pred.
// ---------------------------------------------------------------------------
__global__ __launch_bounds__(256)
void argmax_kernel(const float* __restrict__ scores,
                   const int* __restrict__ targets,
                   int* __restrict__ meta)
{
    int i = blockIdx.x * blockDim.x + threadIdx.x;
    if (i >= NROW) return;
    const float* s = scores + (size_t)i * NCLS;
    float best = s[0];
    int bi = 0;
    for (int j = 1; j < NCLS; ++j) {
        float v = s[j];
        if (v > best) { best = v; bi = j; }
    }
    int t = targets[i];
    meta[2 * i]     = (t != bi) ? t : -1;
    meta[2 * i + 1] = bi;
}

// ---------------------------------------------------------------------------
// Kernel 3: per-class segment sums of misclassified rows + clipped update.
// Each block owns CH columns; add/sub accumulators live in LDS (51.2 KB).
// 8 waves per block: 4 row-groups stride the 16384 rows and accumulate into
// the shared tile with ds_add_f32 (no-return LDS float atomic) -> no RMW
// dependency chain, no bank conflicts (consecutive tid -> consecutive bank).
// encoded is streamed exactly once across all blocks.
// ---------------------------------------------------------------------------
__global__ __launch_bounds__(UPD_THREADS)
void update_kernel(const float* __restrict__ enc,
                   const float* __restrict__ hvs,
                   const int* __restrict__ meta,
                   float* __restrict__ out_hvs)
{
    __shared__ float sAdd[NCLS * CH];
    __shared__ float sSub[NCLS * CH];
    const int tid  = threadIdx.x;
    const int cidx = tid & (CH - 1);           // column within tile
    const int rg   = tid >> 6;                 // row-group 0..3
    const int col  = blockIdx.x * CH + cidx;
    const bool inb = (col < DDIM);

    for (int i = tid; i < NCLS * CH; i += UPD_THREADS) {
        sAdd[i] = 0.0f;
        sSub[i] = 0.0f;
    }
    __syncthreads();

#pragma unroll 4
    for (int r = rg; r < NROW; r += ROWGROUPS) {
        int t = meta[2 * r];                   // uniform per 64-thread group
        if (t >= 0) {
            int p = meta[2 * r + 1];
            float v = inb ? enc[(size_t)r * DDIM + col] : 0.0f;
            atomicAdd(&sAdd[t * CH + cidx], v);   // ds_add_f32
            atomicAdd(&sSub[p * CH + cidx], v);   // ds_add_f32
        }
    }
    __syncthreads();

    const float alpha = 1.0f;
    for (int idx = tid; idx < NCLS * CH; idx += UPD_THREADS) {
        int c = idx / CH;
        int j = idx - c * CH;
        int ocol = blockIdx.x * CH + j;
        if (ocol < DDIM) {
            // torch-faithful: add clipped AFTER *alpha, sub clipped BEFORE *alpha
            float a = fminf(fmaxf(sAdd[idx] * alpha, -1.0f), 1.0f);
            float s = fminf(fmaxf(sSub[idx], -1.0f), 1.0f);
            out_hvs[(size_t)c * DDIM + ocol] =
                hvs[(size_t)c * DDIM + ocol] + a - s * alpha;
        }
    }
}

// ---------------------------------------------------------------------------
extern "C" void kernel_launch(void* const* d_in, const int* in_sizes, int n_in,
                              void* d_out, int out_size, void* d_ws, size_t ws_size,
                              hipStream_t stream)
{
    const float* enc     = (const float*)d_in[0];   // [N, D] fp32
    const int*   targets = (const int*)d_in[1];     // [N] int
    const float* hvs     = (const float*)d_in[2];   // [NCLS, D] fp32

    float* scores  = (float*)d_out;                       // [N, NCLS]
    float* out_hvs = scores + (size_t)NROW * NCLS;        // [NCLS, D]
    int*   meta    = (int*)d_ws;                          // [N, 2]

    gemm_scores_kernel<<<NROW / 128, 256, 0, stream>>>(enc, hvs, scores);
    argmax_kernel<<<(NROW + 255) / 256, 256, 0, stream>>>(scores, targets, meta);
    update_kernel<<<(DDIM + CH - 1) / CH, CH == 64 ? UPD_THREADS : UPD_THREADS,
                    0, stream>>>(enc, hvs, meta, out_hvs);
}